// GraphConv_65652870087417
// MI455X (gfx1250) — compile-verified
//
#include <hip/hip_runtime.h>

typedef __attribute__((ext_vector_type(2))) float v2f;
typedef __attribute__((ext_vector_type(8))) float v8f;

#define N_NODES 50000
#define N_EDGES 800000
#define D_IN    512
#define D_OUT   128
#define N_BANDS (N_NODES / 16)   // 3125 exactly

// --------------------------------------------------------------------------
// Kernel 1: h[M,128] = x[M,512] @ W[512,128] via V_WMMA_F32_16X16X4_F32.
// One wave computes a 16-row band across all N=128 (8 accumulator frags),
// so each x row is streamed from HBM exactly once. W stays L2-resident.
// --------------------------------------------------------------------------
__global__ __launch_bounds__(256) void gemm_wmma_f32(
    const float* __restrict__ x, const float* __restrict__ W,
    float* __restrict__ h)
{
    const int lane = threadIdx.x & 31;
    const int band = blockIdx.x * 8 + (threadIdx.x >> 5);
    if (band >= N_BANDS) return;           // wave-uniform: EXEC stays all-1s

    const int m0   = band * 16;
    const int half = lane >> 4;            // 0 or 1
    const int l16  = lane & 15;
    const int koff = half * 2;             // K-pair offset for A and B frags

    v8f acc[8] = {};                       // 16x128 of C, 8 frags

    // A-frag: lane holds x[m0 + l16, k0+koff .. k0+koff+1]  (ISA 16x4 f32 layout)
    const float* xrow = x + (size_t)(m0 + l16) * D_IN + koff;

    for (int k0 = 0; k0 < D_IN; k0 += 4) {
        v2f a;
        a.x = xrow[k0 + 0];
        a.y = xrow[k0 + 1];
        const float* wk = W + (size_t)(k0 + koff) * D_OUT + l16;
        #pragma unroll
        for (int nt = 0; nt < 8; ++nt) {
            // B-frag: lane holds W[k0+koff, nt*16+l16] and W[k0+koff+1, ...]
            v2f b;
            b.x = wk[nt * 16];
            b.y = wk[nt * 16 + D_OUT];
            acc[nt] = __builtin_amdgcn_wmma_f32_16x16x4_f32(
                /*neg_a=*/false, a, /*neg_b=*/false, b,
                /*c_mod=*/(short)0, acc[nt],
                /*reuse_a=*/false, /*reuse_b=*/false);
        }
    }

    // C/D layout: VGPR r -> row m0 + r + 8*half, col nt*16 + l16
    #pragma unroll
    for (int nt = 0; nt < 8; ++nt) {
        float* hp = h + (size_t)(m0 + half * 8) * D_OUT + nt * 16 + l16;
        #pragma unroll
        for (int r = 0; r < 8; ++r)
            hp[(size_t)r * D_OUT] = acc[nt][r];
    }
}

// --------------------------------------------------------------------------
// Kernel 2: out[i, :] = bias  (initializes the scatter accumulators)
// --------------------------------------------------------------------------
__global__ __launch_bounds__(256) void bias_init(
    const float* __restrict__ b, float* __restrict__ out)
{
    const int idx = blockIdx.x * 256 + threadIdx.x;
    if (idx < N_NODES * D_OUT)
        out[idx] = b[idx & (D_OUT - 1)];
}

// --------------------------------------------------------------------------
// Kernel 3: out[dst[e], :] += edge_w[e] * h[src[e], :]
// One wave per edge: coalesced 512B gather of h (L2-resident), then
// 4 native global_atomic_add_f32 per lane into out (also L2-resident).
// --------------------------------------------------------------------------
__global__ __launch_bounds__(256) void edge_scatter(
    const float* __restrict__ h, const float* __restrict__ edge_w,
    const int* __restrict__ src, const int* __restrict__ dst,
    float* __restrict__ out)
{
    const int gtid = blockIdx.x * 256 + threadIdx.x;
    const int e    = gtid >> 5;
    const int lane = threadIdx.x & 31;
    if (e >= N_EDGES) return;

    const int   s = src[e];
    const int   d = dst[e];
    const float w = edge_w[e];

    const float4 hv = *(const float4*)(h + (size_t)s * D_OUT + lane * 4);
    float* op = out + (size_t)d * D_OUT + lane * 4;

    __hip_atomic_fetch_add(op + 0, w * hv.x, __ATOMIC_RELAXED, __HIP_MEMORY_SCOPE_AGENT);
    __hip_atomic_fetch_add(op + 1, w * hv.y, __ATOMIC_RELAXED, __HIP_MEMORY_SCOPE_AGENT);
    __hip_atomic_fetch_add(op + 2, w * hv.z, __ATOMIC_RELAXED, __HIP_MEMORY_SCOPE_AGENT);
    __hip_atomic_fetch_add(op + 3, w * hv.w, __ATOMIC_RELAXED, __HIP_MEMORY_SCOPE_AGENT);
}

// --------------------------------------------------------------------------
extern "C" void kernel_launch(void* const* d_in, const int* in_sizes, int n_in,
                              void* d_out, int out_size, void* d_ws, size_t ws_size,
                              hipStream_t stream)
{
    const float* x      = (const float*)d_in[0];
    const float* edge_w = (const float*)d_in[1];
    const int*   src    = (const int*)  d_in[2];
    const int*   dst    = (const int*)  d_in[3];
    const float* W      = (const float*)d_in[4];
    const float* b      = (const float*)d_in[5];
    float*       out    = (float*)d_out;
    float*       h      = (float*)d_ws;   // 50000*128*4 = 25.6 MB scratch

    // 1) GEMM: 3125 bands, 8 waves (bands) per 256-thread block
    gemm_wmma_f32<<<(N_BANDS + 7) / 8, 256, 0, stream>>>(x, W, h);

    // 2) out = bias (broadcast)
    bias_init<<<(N_NODES * D_OUT + 255) / 256, 256, 0, stream>>>(b, out);

    // 3) scatter-add: one wave per edge
    edge_scatter<<<(N_EDGES * 32) / 256, 256, 0, stream>>>(h, edge_w, src, dst, out);
}